// VectorQuantizer_747324309715
// MI455X (gfx1250) — compile-verified
//
#include <hip/hip_runtime.h>

typedef __attribute__((ext_vector_type(2))) float v2f;
typedef __attribute__((ext_vector_type(8))) float v8f;

constexpr int N_ROWS  = 32 * 32 * 32;  // 32768 rows of x
constexpr int DIM     = 64;            // embedding dim
constexpr int K_CODES = 1024;          // codebook size
constexpr int GROUP_CODES = 128;       // codes staged in LDS per chunk
constexpr int LDS_STRIDE  = 136;       // 128 + 8 pad: lanes 0-15 / 16-31 hit disjoint banks

// ---------------------------------------------------------------------------
// Kernel 1: per-code squared norms + zero the loss accumulator.
// ---------------------------------------------------------------------------
__global__ void vq_prep(const float* __restrict__ emb,
                        float* __restrict__ eNorm,
                        float* __restrict__ lossAcc) {
    int k = blockIdx.x * blockDim.x + threadIdx.x;
    if (k < K_CODES) {
        float s = 0.f;
        #pragma unroll 8
        for (int d = 0; d < DIM; ++d) {
            float e = emb[d * K_CODES + k];
            s += e * e;
        }
        eNorm[k] = s;
    }
    if (k == 0) *lossAcc = 0.f;
}

// ---------------------------------------------------------------------------
// Kernel 2: fused GEMM + argmin using V_WMMA_F32_16X16X4_F32.
// Each wave owns a 16-row tile of x; block stages 128 codes at a time in LDS.
// argmin_k(||x||^2 + ||e_k||^2 - 2 sim)  ==  argmin_k(||e_k||^2 - 2 sim).
// All 16 B-fragment DS loads are issued before the WMMA chain (enforced with
// a sched_barrier) so LDS latency overlaps the XDL work; accumulation runs
// as two independent 8-deep chains for ILP on the D->C dependence.
// ---------------------------------------------------------------------------
__global__ __launch_bounds__(256) void vq_argmin(const float* __restrict__ x,
                                                 const float* __restrict__ emb,
                                                 const float* __restrict__ eNorm,
                                                 int* __restrict__ idxOut) {
    __shared__ float lds_e[DIM * LDS_STRIDE];   // ~34 KB

    const int tid    = threadIdx.x;
    const int lane   = tid & 31;
    const int waveId = tid >> 5;
    const int lo     = lane & 15;   // column / row-in-tile index
    const int hi     = lane >> 4;   // selects K pair (A/B) and M half (C/D)

    const int rowBase = blockIdx.x * 128 + waveId * 16;
    const float* xrow = x + (rowBase + lo) * DIM;

    // A fragments: 16 fragments cover K = 0..63 (4 per WMMA).
    // Fragment s: VGPR0 = K=4s (+2 for hi lanes), VGPR1 = K=4s+1 (+2).
    v2f a[16];
    #pragma unroll
    for (int s = 0; s < 16; ++s)
        a[s] = *reinterpret_cast<const v2f*>(xrow + 4 * s + 2 * hi);

    float bestV[8];
    int   bestI[8];
    #pragma unroll
    for (int r = 0; r < 8; ++r) { bestV[r] = 3.4e38f; bestI[r] = 0; }

    for (int g = 0; g < 8; ++g) {
        const int kBase = g * GROUP_CODES;
        // Stage 64 dims x 128 codes into LDS (padded stride), all 256 threads.
        #pragma unroll 4
        for (int i = 0; i < 32; ++i) {
            int lin = i * 256 + tid;
            int d   = lin >> 7;
            int c   = lin & 127;
            lds_e[d * LDS_STRIDE + c] = emb[d * K_CODES + kBase + c];
        }
        __syncthreads();

        for (int tl = 0; tl < 8; ++tl) {
            const int col = kBase + tl * 16 + lo;  // this lane's column in C
            const float* bp = &lds_e[tl * 16 + lo];

            // Prefetch all 16 B fragments; keep every ds_load in flight.
            v2f b[16];
            #pragma unroll
            for (int s = 0; s < 16; ++s) {
                b[s].x = bp[(4 * s + 2 * hi) * LDS_STRIDE];
                b[s].y = bp[(4 * s + 2 * hi + 1) * LDS_STRIDE];
            }
#if __has_builtin(__builtin_amdgcn_sched_barrier)
            // Do not let the scheduler sink the DS loads into the WMMA chain.
            __builtin_amdgcn_sched_barrier(0);
#endif

            // Two independent accumulation chains for ILP on the D->C dep.
            v8f acc0 = {0.f, 0.f, 0.f, 0.f, 0.f, 0.f, 0.f, 0.f};
            v8f acc1 = {0.f, 0.f, 0.f, 0.f, 0.f, 0.f, 0.f, 0.f};
            #pragma unroll
            for (int s = 0; s < 8; ++s) {
                acc0 = __builtin_amdgcn_wmma_f32_16x16x4_f32(
                           false, a[2 * s],     false, b[2 * s],     (short)0, acc0, false, false);
                acc1 = __builtin_amdgcn_wmma_f32_16x16x4_f32(
                           false, a[2 * s + 1], false, b[2 * s + 1], (short)0, acc1, false, false);
            }

            const float en = eNorm[col];
            #pragma unroll
            for (int r = 0; r < 8; ++r) {
                float sim = acc0[r] + acc1[r];
                float d2  = en - 2.0f * sim;        // row M = r + 8*hi
                if (d2 < bestV[r]) { bestV[r] = d2; bestI[r] = col; }
            }
        }
        __syncthreads();
    }

    // Min-reduce across the 16 lanes holding the same row (xor < 16 stays
    // inside each half-wave). Tie-break on smaller index like jnp.argmin.
    #pragma unroll
    for (int r = 0; r < 8; ++r) {
        float bv = bestV[r];
        int   bi = bestI[r];
        #pragma unroll
        for (int off = 8; off >= 1; off >>= 1) {
            float ov = __shfl_xor(bv, off, 32);
            int   oi = __shfl_xor(bi, off, 32);
            if (ov < bv || (ov == bv && oi < bi)) { bv = ov; bi = oi; }
        }
        if (lo == 0) idxOut[rowBase + hi * 8 + r] = bi;
    }
}

// ---------------------------------------------------------------------------
// Kernel 3: gather quantized vectors (== straight-through output) and
// accumulate sum((q - x)^2) via block reduction + one atomic per block.
// ---------------------------------------------------------------------------
__global__ __launch_bounds__(256) void vq_gather_loss(const float* __restrict__ x,
                                                      const float* __restrict__ emb,
                                                      const int* __restrict__ idx,
                                                      float* __restrict__ out,
                                                      float* __restrict__ lossAcc) {
    __shared__ float red[256];
    const int tid = threadIdx.x;
    const int e0  = (blockIdx.x * 256 + tid) * 4;   // 4 contiguous dims, same row
    const int row = e0 >> 6;
    const int d0  = e0 & 63;
    const int id  = idx[row];

    float q0 = emb[(d0 + 0) * K_CODES + id];
    float q1 = emb[(d0 + 1) * K_CODES + id];
    float q2 = emb[(d0 + 2) * K_CODES + id];
    float q3 = emb[(d0 + 3) * K_CODES + id];

    float4 xv = *reinterpret_cast<const float4*>(x + e0);
    float t0 = q0 - xv.x, t1 = q1 - xv.y, t2 = q2 - xv.z, t3 = q3 - xv.w;
    float lsum = t0 * t0 + t1 * t1 + t2 * t2 + t3 * t3;

    float4 qv = make_float4(q0, q1, q2, q3);
    *reinterpret_cast<float4*>(out + e0) = qv;

    red[tid] = lsum;
    __syncthreads();
    #pragma unroll
    for (int s = 128; s > 0; s >>= 1) {
        if (tid < s) red[tid] += red[tid + s];
        __syncthreads();
    }
    if (tid == 0) atomicAdd(lossAcc, red[0]);
}

// ---------------------------------------------------------------------------
// Kernel 4: loss = (beta + 1) * mean = 1.25 * sum / (N*D)
// ---------------------------------------------------------------------------
__global__ void vq_finalize(const float* __restrict__ lossAcc,
                            float* __restrict__ lossOut) {
    *lossOut = 1.25f * (*lossAcc) / (float)(N_ROWS * DIM);
}

// ---------------------------------------------------------------------------
extern "C" void kernel_launch(void* const* d_in, const int* in_sizes, int n_in,
                              void* d_out, int out_size, void* d_ws, size_t ws_size,
                              hipStream_t stream) {
    const float* x   = (const float*)d_in[0];   // (32,32,32,64) fp32
    const float* emb = (const float*)d_in[1];   // (64,1024) fp32
    float* out = (float*)d_out;                 // 32768*64 outputs + 1 loss

    char*  ws      = (char*)d_ws;
    float* lossAcc = (float*)ws;                // 1 float
    float* eNorm   = (float*)(ws + 64);         // 1024 floats
    int*   idx     = (int*)(ws + 4352);         // 32768 ints (256B-aligned)

    vq_prep       <<<(K_CODES + 255) / 256, 256, 0, stream>>>(emb, eNorm, lossAcc);
    vq_argmin     <<<N_ROWS / 128, 256, 0, stream>>>(x, emb, eNorm, idx);
    vq_gather_loss<<<(N_ROWS * DIM) / (256 * 4), 256, 0, stream>>>(x, emb, idx, out, lossAcc);
    vq_finalize   <<<1, 1, 0, stream>>>(lossAcc, out + (size_t)N_ROWS * DIM);
}